// MotionPrimitiveDecoder_83451214561465
// MI455X (gfx1250) — compile-verified
//
#include <hip/hip_runtime.h>
#include <hip/hip_bf16.h>

typedef __attribute__((ext_vector_type(2))) float v2f;
typedef __attribute__((ext_vector_type(8))) float v8f;

#define B_  32
#define M_  512
#define L_  128
#define N_  64
#define T_  40
#define A_  6
#define Z_  64
#define TA_ 240          // T_*A_
#define EPS_ 1e-8f
#define S_SCALE_ 50.0f
#define D_SCALE_ 3.6f

// One workgroup (256 threads = 8 wave32) per (b,n) point.
//   Phase A: polyline projection -> frenet -> per-point ctx constant
//   Phase B: u = dot(ctx,w)+const, masked mean over A, fused bias s_ub
//   Phase C: WMMA f32 16x16x4 batched matvec (240x64 @ 64) + fused bias writeout
__launch_bounds__(256, 2)
__global__ void MotionPrimitiveDecoder_83451214561465_kernel(
    const float* __restrict__ mapp,   // (B,M,L,2)
    const int*   __restrict__ idx,    // (B,N)
    const float* __restrict__ pts,    // (B,N,2)
    const float* __restrict__ z,      // (B,N,Z)
    const float* __restrict__ dec,    // (B,N,T,A,Z)
    const float* __restrict__ ctx,    // (B,N,T,A,3)
    const int*   __restrict__ fa,     // (B,N,T,A)
    const float* __restrict__ ucw,    // (3,)
    const float* __restrict__ ucb,    // (1,)
    float*       __restrict__ out)    // (B,N,T,A)
{
    __shared__ float s_len[L_];   // segment lengths (idx 127 = 0)
    __shared__ float s_scan[L_];  // inclusive scan buffer
    __shared__ float s_cum[L_];   // exclusive prefix (cum0)
    __shared__ float s_t[L_];
    __shared__ float s_vx[L_];
    __shared__ float s_vy[L_];
    __shared__ float s_dx[L_];
    __shared__ float s_dy[L_];
    __shared__ float s_d2[L_];    // argmin reduce: min dist2
    __shared__ int   s_ix[L_];    // argmin reduce: index
    __shared__ float s_z[Z_];
    __shared__ float s_u[TA_];
    __shared__ int   s_fa[TA_];
    __shared__ float s_ub[TA_];   // fused bias: u - u_mean (no div in hot tail)
    __shared__ float s_cbn;       // dot(frenet, w) + bias, constant per (b,n)

    const int n   = blockIdx.x;
    const int b   = blockIdx.y;
    const int tid = threadIdx.x;
    const int bn  = b * N_ + n;

    // stage z for this (b,n) into LDS (feeds WMMA B fragments later)
    if (tid < Z_) s_z[tid] = z[(size_t)bn * Z_ + tid];

    // ---------------- Phase A: project point onto gathered polyline ----------
    float seg_len = 0.0f;
    if (tid < L_ - 1) {
        const int   pid = idx[bn];
        const float* pb = mapp + ((size_t)(b * M_ + pid) * L_) * 2;
        const float p0x = pb[2 * tid],     p0y = pb[2 * tid + 1];
        const float p1x = pb[2 * tid + 2], p1y = pb[2 * tid + 3];
        const float px  = pts[(size_t)bn * 2];
        const float py  = pts[(size_t)bn * 2 + 1];
        const float vx = p1x - p0x, vy = p1y - p0y;
        const float v2 = fmaxf(vx * vx + vy * vy, EPS_);
        seg_len = sqrtf(v2);
        const float wx = px - p0x, wy = py - p0y;
        const float tt = fminf(fmaxf((wx * vx + wy * vy) / v2, 0.0f), 1.0f);
        const float dx = wx - tt * vx, dy = wy - tt * vy;
        s_len[tid] = seg_len; s_t[tid] = tt;
        s_vx[tid] = vx; s_vy[tid] = vy;
        s_dx[tid] = dx; s_dy[tid] = dy;
        s_d2[tid] = dx * dx + dy * dy;
        s_ix[tid] = tid;
    } else if (tid < L_) {
        s_len[tid] = 0.0f;
        s_d2[tid]  = 3.4e38f;   // never wins argmin
        s_ix[tid]  = tid;
    }
    if (tid < L_) s_scan[tid] = seg_len;
    __syncthreads();

    // inclusive Hillis-Steele scan of segment lengths (128 slots)
    #pragma unroll
    for (int off = 1; off < L_; off <<= 1) {
        float add = 0.0f;
        if (tid < L_ && tid >= off) add = s_scan[tid - off];
        __syncthreads();
        if (tid < L_) s_scan[tid] += add;
        __syncthreads();
    }
    if (tid < L_) s_cum[tid] = s_scan[tid] - s_len[tid];  // exclusive prefix

    // argmin over dist2 (tie -> smaller segment index, matching jnp.argmin)
    for (int str = L_ / 2; str > 0; str >>= 1) {
        __syncthreads();
        if (tid < str) {
            const float da = s_d2[tid],       db = s_d2[tid + str];
            const int   ia = s_ix[tid];
            const int   ib = s_ix[tid + str];
            if (db < da || (db == da && ib < ia)) { s_d2[tid] = db; s_ix[tid] = ib; }
        }
    }
    __syncthreads();
    if (tid == 0) {
        const int   j    = s_ix[0];
        const float d2s  = fmaxf(s_d2[0], EPS_);
        const float s    = s_cum[j] + s_t[j] * s_len[j];
        const float cross = s_vx[j] * s_dy[j] - s_vy[j] * s_dx[j];
        const float sign  = (cross > 0.0f) ? 1.0f : ((cross < 0.0f) ? -1.0f : 1.0f);
        const float d     = sign * sqrtf(d2s);
        const float tanx  = s_vx[j] / fmaxf(s_len[j], EPS_);
        const float f0 = s / S_SCALE_, f1 = d / D_SCALE_, f2 = tanx;
        s_cbn = f0 * ucw[0] + f1 * ucw[1] + f2 * ucw[2] + ucb[0];
    }
    __syncthreads();

    // ---------------- Phase B: utility term + masked mean over actions -------
    if (tid < TA_) {
        const size_t base = (size_t)bn * TA_ + tid;
        const float* cp = ctx + base * 3;
        s_u[tid]  = cp[0] * ucw[0] + cp[1] * ucw[1] + cp[2] * ucw[2] + s_cbn;
        s_fa[tid] = fa[base];
    }
    __syncthreads();
    if (tid < T_) {
        float cnt = 0.0f, su = 0.0f, sa = 0.0f;
        #pragma unroll
        for (int a = 0; a < A_; ++a) {
            const float uv = s_u[tid * A_ + a];
            sa += uv;
            if (s_fa[tid * A_ + a]) { cnt += 1.0f; su += uv; }
        }
        const float um = (cnt > 0.0f) ? (su / cnt) : (sa * (1.0f / 6.0f));
        #pragma unroll
        for (int a = 0; a < A_; ++a)
            s_ub[tid * A_ + a] = s_u[tid * A_ + a] - um;  // fused bias, no /6 later
    }
    __syncthreads();

    // ---------------- Phase C: WMMA matvec logits = dec(240x64) @ z(64) ------
    // A fragment (16x4 f32): lane&15 = row M, lane half selects K pair, 2 VGPRs.
    // B fragment: z K-chunk replicated across all 16 N columns (redundant
    // columns are free — kernel is HBM-bound at ~1 FLOP/byte).
    const int wave = tid >> 5;
    const int lane = tid & 31;
    const int koff = (lane >> 4) << 1;                 // 0 for lanes 0-15, 2 for 16-31
    const float* dbase = dec + (size_t)bn * TA_ * Z_;

    for (int tile = wave; tile < TA_ / 16; tile += 8) {
        const int row = tile * 16 + (lane & 15);
        const float* ar = dbase + (size_t)row * Z_;
        v8f acc = {0.f, 0.f, 0.f, 0.f, 0.f, 0.f, 0.f, 0.f};
        #pragma unroll
        for (int kc = 0; kc < Z_ / 4; ++kc) {
            const int k = kc * 4 + koff;
            v2f av; av.x = ar[k];  av.y = ar[k + 1];    // global_load_b64 per lane
            v2f bv; bv.x = s_z[k]; bv.y = s_z[k + 1];   // ds_load (hoisted pre-loop)
            acc = __builtin_amdgcn_wmma_f32_16x16x4_f32(
                false, av, false, bv, (short)0, acc, false, false);
        }
        // C/D layout: lanes 0-15 hold M = 0..7 in VGPR 0..7 (all N identical),
        // lanes 16-31 hold M = 8..15. Lane 0 / lane 16 write 8 rows each.
        if ((lane & 15) == 0) {
            const int mbase = tile * 16 + ((lane >> 4) << 3);
            float* op = out + (size_t)bn * TA_;
            #pragma unroll
            for (int e = 0; e < 8; ++e) {
                const int r = mbase + e;
                op[r] = acc[e] + s_ub[r];
            }
        }
    }
}

extern "C" void kernel_launch(void* const* d_in, const int* in_sizes, int n_in,
                              void* d_out, int out_size, void* d_ws, size_t ws_size,
                              hipStream_t stream) {
    (void)in_sizes; (void)n_in; (void)out_size; (void)d_ws; (void)ws_size;
    const float* mapp = (const float*)d_in[0];
    const int*   idx  = (const int*)  d_in[1];
    const float* pts  = (const float*)d_in[2];
    const float* z    = (const float*)d_in[3];
    const float* dec  = (const float*)d_in[4];
    const float* ctx  = (const float*)d_in[5];
    const int*   fa   = (const int*)  d_in[6];
    const float* ucw  = (const float*)d_in[7];
    const float* ucb  = (const float*)d_in[8];
    float* out = (float*)d_out;

    dim3 grid(N_, B_);
    dim3 block(256);
    MotionPrimitiveDecoder_83451214561465_kernel<<<grid, block, 0, stream>>>(
        mapp, idx, pts, z, dec, ctx, fa, ucw, ucb, out);
}